// MoEGate_82437602279913
// MI455X (gfx1250) — compile-verified
//
#include <hip/hip_runtime.h>
#include <hip/hip_bf16.h>
#include <math.h>

typedef __attribute__((ext_vector_type(2))) float v2f;
typedef __attribute__((ext_vector_type(8))) float v8f;

#define HID      1024
#define NE       64
#define TOPK     8
#define SEQ      4096
#define NBATCH   8
#define TOK_PER_BLK 64

// ---------------------------------------------------------------------------
// Kernel 0: zero the per-(batch,expert) workspace accumulators.
// ws layout: int cnt_be[8][64] | float ssum_be[8][64]  => 1024 dwords
// ---------------------------------------------------------------------------
__global__ void __launch_bounds__(256) moe_zero_ws(int* __restrict__ ws) {
    for (int i = threadIdx.x; i < 1024; i += 256) ws[i] = 0;
}

// ---------------------------------------------------------------------------
// Kernel 1: fused gate. Per block: 64 tokens.
//  Phase A: WMMA f32 16x16x4 GEMM  logits[64][64] -> LDS
//  Phase B: per-token softmax, column score sums, top-8, counts.
// ---------------------------------------------------------------------------
__global__ void __launch_bounds__(128) moe_gate_kernel(
    const float* __restrict__ hidden,   // [32768,1024]
    const float* __restrict__ weight,   // [64,1024]
    int*   __restrict__ idx_out,        // [32768,8]
    float* __restrict__ w_out,          // [32768,8]
    int*   __restrict__ cnt_be,         // [8,64]
    float* __restrict__ ssum_be)        // [8,64]
{
    __shared__ float lds[TOK_PER_BLK][NE + 1];   // logits -> scores (in place)
    __shared__ int   cnt[NE];

    const int tid     = threadIdx.x;
    const int lane    = tid & 31;
    const int wave    = tid >> 5;
    const int tokBase = blockIdx.x * TOK_PER_BLK;
    const int b       = tokBase / SEQ;           // batch row (block never straddles)

    if (tid < NE) cnt[tid] = 0;

    // ---------------- Phase A: WMMA GEMM ----------------
    // A 16x4 f32 fragment: lane L holds row M=L%16, K-pair offset (L/16)*2
    // B 4x16 fragment:     lane L holds col N=L%16, same K-pair offset
    const int m    = lane & 15;
    const int koff = (lane >> 4) << 1;

    const float* aRow = hidden + (size_t)(tokBase + wave * 16 + m) * HID + koff;
    const float* bp0  = weight + (size_t)( 0 + m) * HID + koff;
    const float* bp1  = weight + (size_t)(16 + m) * HID + koff;
    const float* bp2  = weight + (size_t)(32 + m) * HID + koff;
    const float* bp3  = weight + (size_t)(48 + m) * HID + koff;

    v8f acc0 = {}, acc1 = {}, acc2 = {}, acc3 = {};
#pragma unroll 4
    for (int k0 = 0; k0 < HID; k0 += 4) {
        v2f a  = *(const v2f*)(aRow + k0);
        v2f w0 = *(const v2f*)(bp0 + k0);
        v2f w1 = *(const v2f*)(bp1 + k0);
        v2f w2 = *(const v2f*)(bp2 + k0);
        v2f w3 = *(const v2f*)(bp3 + k0);
        acc0 = __builtin_amdgcn_wmma_f32_16x16x4_f32(false, a, false, w0, (short)0, acc0, false, false);
        acc1 = __builtin_amdgcn_wmma_f32_16x16x4_f32(false, a, false, w1, (short)0, acc1, false, false);
        acc2 = __builtin_amdgcn_wmma_f32_16x16x4_f32(false, a, false, w2, (short)0, acc2, false, false);
        acc3 = __builtin_amdgcn_wmma_f32_16x16x4_f32(false, a, false, w3, (short)0, acc3, false, false);
    }

    // C/D layout: VGPR r, lanes 0-15 -> M=r, N=lane; lanes 16-31 -> M=r+8, N=lane-16
    const int rowBase = wave * 16 + ((lane >> 4) << 3);
    const int col     = lane & 15;
#pragma unroll
    for (int r = 0; r < 8; ++r) {
        lds[rowBase + r][col     ] = acc0[r];
        lds[rowBase + r][col + 16] = acc1[r];
        lds[rowBase + r][col + 32] = acc2[r];
        lds[rowBase + r][col + 48] = acc3[r];
    }
    __syncthreads();

    // ---------------- Phase B1: softmax per token (in place) ----------------
    if (tid < TOK_PER_BLK) {
        float* row = lds[tid];
        float mx = -3.4e38f;
        for (int e = 0; e < NE; ++e) mx = fmaxf(mx, row[e]);
        float denom = 0.f;
        for (int e = 0; e < NE; ++e) { float s = expf(row[e] - mx); denom += s; row[e] = s; }
        float inv = 1.0f / denom;
        for (int e = 0; e < NE; ++e) row[e] *= inv;
    }
    __syncthreads();

    // ---------------- Phase B2: per-expert score sums (aux loss) ------------
    if (tid < NE) {
        float s = 0.f;
        for (int t = 0; t < TOK_PER_BLK; ++t) s += lds[t][tid];
        atomicAdd(&ssum_be[b * NE + tid], s);
    }
    __syncthreads();

    // ---------------- Phase B3: top-8, renormalize, counts ------------------
    if (tid < TOK_PER_BLK) {
        const int token = tokBase + tid;
        float* row = lds[tid];
        float wk[TOPK]; int ik[TOPK];
        float topsum = 0.f;
#pragma unroll
        for (int r = 0; r < TOPK; ++r) {
            float best = -1.f; int bi = 0;
            for (int e = 0; e < NE; ++e) {
                float v = row[e];
                if (v > best) { best = v; bi = e; }   // strict > : lowest index wins ties
            }
            row[bi] = -2.f;
            wk[r] = best; ik[r] = bi; topsum += best;
            atomicAdd(&cnt[bi], 1);
        }
        const float invTop = 1.0f / (topsum + 1e-20f);
#pragma unroll
        for (int r = 0; r < TOPK; ++r) {
            idx_out[(size_t)token * TOPK + r] = ik[r];
            w_out  [(size_t)token * TOPK + r] = wk[r] * invTop;
        }
    }
    __syncthreads();

    if (tid < NE) atomicAdd(&cnt_be[b * NE + tid], cnt[tid]);
}

// ---------------------------------------------------------------------------
// Kernel 2: reduce workspace -> expert_counts + aux_loss scalar.
// aux = ALPHA * mean_b( sum_e  cnt[b,e]*(NE/(SEQ*TOPK)) * ssum[b,e]/SEQ )
// ---------------------------------------------------------------------------
__global__ void __launch_bounds__(64) moe_finalize_kernel(
    const int*   __restrict__ cnt_be,
    const float* __restrict__ ssum_be,
    float* __restrict__ aux_out,
    int*   __restrict__ counts_out)
{
    __shared__ float red[NE];
    const int e = threadIdx.x;
    int   total = 0;
    float part  = 0.f;
    const float ceScale = (float)NE / (float)(SEQ * TOPK);
    const float sScale  = 1.0f / (float)SEQ;
    for (int b = 0; b < NBATCH; ++b) {
        int c = cnt_be[b * NE + e];
        total += c;
        part  += (float)c * ceScale * (ssum_be[b * NE + e] * sScale);
    }
    counts_out[e] = total;
    red[e] = part;
    __syncthreads();
    for (int s = 32; s > 0; s >>= 1) {
        if (e < s) red[e] += red[e + s];
        __syncthreads();
    }
    if (e == 0) *aux_out = 0.01f * red[0] / (float)NBATCH;
}

// ---------------------------------------------------------------------------
extern "C" void kernel_launch(void* const* d_in, const int* in_sizes, int n_in,
                              void* d_out, int out_size, void* d_ws, size_t ws_size,
                              hipStream_t stream) {
    const float* hidden = (const float*)d_in[0];   // [8,4096,1024] f32
    const float* weight = (const float*)d_in[1];   // [64,1024] f32

    const int T = 32768;
    int*   cnt_be  = (int*)d_ws;                               // [8][64]
    float* ssum_be = (float*)((char*)d_ws + 512 * sizeof(int));// [8][64]

    int*   idx_out    = (int*)d_out;                           // [T*8] int32
    float* w_out      = (float*)d_out + (size_t)T * TOPK;      // [T*8] f32
    float* aux_out    = (float*)d_out + (size_t)2 * T * TOPK;  // [1]   f32
    int*   counts_out = (int*)d_out + (size_t)2 * T * TOPK + 1;// [64]  int32

    moe_zero_ws<<<1, 256, 0, stream>>>((int*)d_ws);
    moe_gate_kernel<<<T / TOK_PER_BLK, 128, 0, stream>>>(
        hidden, weight, idx_out, w_out, cnt_be, ssum_be);
    moe_finalize_kernel<<<1, 64, 0, stream>>>(cnt_be, ssum_be, aux_out, counts_out);
}